// DNCSolver_77713138254395
// MI455X (gfx1250) — compile-verified
//
#include <hip/hip_runtime.h>
#include <hip/hip_bf16.h>
#include <math.h>

// Problem constants (match reference)
#define B_   32
#define T_   128
#define IN_  512
#define OUT_ 512
#define N_   128
#define W_   64
#define R_   4
#define H_   512
#define EPS_ 1e-6f
#define IFACE_   471
#define VSTRIDE_ 480

typedef float v2f_t __attribute__((ext_vector_type(2)));
typedef float v8f_t __attribute__((ext_vector_type(8)));
typedef unsigned int v4u_t __attribute__((ext_vector_type(4)));
typedef int v4i_t __attribute__((ext_vector_type(4)));
typedef int v8i_t __attribute__((ext_vector_type(8)));

__device__ __forceinline__ float sigmoidf_(float x) { return 1.0f / (1.0f + expf(-x)); }
__device__ __forceinline__ float softplusf_(float x) { return (x > 20.0f) ? x : log1pf(expf(x)); }

// ---------------------------------------------------------------------------
// f32 WMMA 16x16x4 wave-level GEMM accumulate: acc += A[m0:m0+16, :K] * B[:K, n0:n0+16]
// A row-major (lda), B row-major (ldb). Unguarded variant: all 16 B columns valid.
// Lane layout per CDNA5 ISA 7.12.2 (f32):
//   A (16x4): lanes 0-15 rows M=0..15; vgpr v = K{v | v+2} by lane half
//   B (4x16): vgpr v = rows K{v | v+2} by lane half, N=lane%16
//   C/D (16x16): vgpr v = row M{v | v+8} by lane half, N=lane%16
// ---------------------------------------------------------------------------
__device__ __forceinline__ v8f_t gemm_wave_full(const float* __restrict__ A, long lda,
                                                const float* __restrict__ B, long ldb,
                                                int K, int m0, int n0, v8f_t acc)
{
  const int lane = threadIdx.x & 31;
  const int half = lane >> 4;
  const int l16  = lane & 15;
  const float* ap = A + (long)(m0 + l16) * lda + 2 * half;
  const float* bp = B + (n0 + l16) + (long)(2 * half) * ldb;
  for (int k = 0; k < K; k += 4) {
    v2f_t a, b;
    const float* a2 = ap + k;
    a[0] = a2[0];
    a[1] = a2[1];
    const float* b2 = bp + (long)k * ldb;
    b[0] = b2[0];
    b[1] = b2[ldb];
    // 8 args: (neg_a, A, neg_b, B, c_mod, C, reuse_a, reuse_b)
    acc = __builtin_amdgcn_wmma_f32_16x16x4_f32(false, a, false, b, (short)0, acc,
                                                false, false);
  }
  return acc;
}

// Guarded variant for ragged N (interface GEMM, N=471): zero-fill OOB B columns.
__device__ __forceinline__ v8f_t gemm_wave_guard(const float* __restrict__ A, long lda,
                                                 const float* __restrict__ B, long ldb,
                                                 int K, int m0, int n0, int nmax, v8f_t acc)
{
  const int lane = threadIdx.x & 31;
  const int half = lane >> 4;
  const int l16  = lane & 15;
  const float* ap = A + (long)(m0 + l16) * lda + 2 * half;
  const bool bok  = (n0 + l16) < nmax;
  const float* bp = B + (n0 + l16) + (long)(2 * half) * ldb;
  for (int k = 0; k < K; k += 4) {
    v2f_t a, b;
    const float* a2 = ap + k;
    a[0] = a2[0];
    a[1] = a2[1];
    if (bok) {
      const float* b2 = bp + (long)k * ldb;
      b[0] = b2[0];
      b[1] = b2[ldb];
    } else {
      b[0] = 0.0f;
      b[1] = 0.0f;
    }
    acc = __builtin_amdgcn_wmma_f32_16x16x4_f32(false, a, false, b, (short)0, acc,
                                                false, false);
  }
  return acc;
}

// ---------------------------------------------------------------------------
// Tensor Data Mover: 2D f32 tile between global and LDS.
// This toolchain exposes the 6-arg builtin form:
//   (uint32x4 g0, int32x8 g1, int32x4 g2, int32x4 g3, int32x8 g4, i32 cpol)
// D# per CDNA5 ISA: group0 = {count|flags, lds_addr, global_addr lo, addr hi|type},
// group1 = {mask/data_size/pad, dims/tiles/stride}; remaining groups zero (2D tensor).
// Issue from one wave only; EXEC is ignored by TDM. Tracked by TENSORcnt.
// ---------------------------------------------------------------------------
__device__ __forceinline__ void tdm_tile_2d_f32(bool store, void* lds_ptr, void* gptr,
                                                unsigned tensor_d0, unsigned tensor_d1,
                                                unsigned tile_d0, unsigned tile_d1,
                                                unsigned stride0_elems)
{
  unsigned lds_off = (unsigned)(unsigned long long)(size_t)lds_ptr; // LDS aperture: addr[31:0]
  unsigned long long ga = (unsigned long long)(size_t)gptr;
  v4u_t g0;
  g0[0] = 1u;                                                   // count=1, user descriptor
  g0[1] = lds_off;                                              // lds_addr [63:32]
  g0[2] = (unsigned)(ga & 0xffffffffu);                         // global_addr [95:64]
  g0[3] = (unsigned)((ga >> 32) & 0x01ffffffu) | (2u << 30);    // addr[56:32] | type=2
  v8i_t g1;
  g1[0] = (int)(2u << 16);                                      // data_size=2 (4 bytes)
  g1[1] = (int)(tensor_d0 << 16);                               // tensor_dim0[15:0]
  g1[2] = (int)((tensor_d0 >> 16) | (tensor_d1 << 16));         // dim0[31:16]|dim1[15:0]
  g1[3] = (int)((tensor_d1 >> 16) | (tile_d0 << 16));           // dim1[31:16]|tile_dim0
  g1[4] = (int)(tile_d1 & 0xffffu);                             // tile_dim1 | tile_dim2=0
  g1[5] = (int)stride0_elems;                                   // tensor_dim0_stride[31:0]
  g1[6] = 0;
  g1[7] = 0;
  v4i_t gz4 = {0, 0, 0, 0};
  v8i_t gz8 = {0, 0, 0, 0, 0, 0, 0, 0};
  if (store)
    __builtin_amdgcn_tensor_store_from_lds(g0, g1, gz4, gz4, gz8, 0);
  else
    __builtin_amdgcn_tensor_load_to_lds(g0, g1, gz4, gz4, gz8, 0);
}

// ---------------------------------------------------------------------------
// Zero-init persistent state in workspace
// ---------------------------------------------------------------------------
__global__ void k_zero(float* __restrict__ p, int n)
{
  int i = blockIdx.x * blockDim.x + threadIdx.x;
  if (i < n) p[i] = 0.0f;
}

// ---------------------------------------------------------------------------
// LSTM step: z = [x_t, reads] @ Wx + h_prev @ Wh + b ; gates -> h_next, c (in place)
// grid (32, 2): blockIdx.x = H col-group (16 cols), blockIdx.y = batch tile (16 rows)
// 128 threads = 4 waves, wave g computes gate g's 16x16 tile.
// ---------------------------------------------------------------------------
__global__ void __launch_bounds__(128) k_lstm(const float* __restrict__ xseq, int t,
                                              const float* __restrict__ Wx,
                                              const float* __restrict__ Wh,
                                              const float* __restrict__ bl,
                                              const float* __restrict__ reads,
                                              const float* __restrict__ h_prev,
                                              float* __restrict__ h_next,
                                              float* __restrict__ c)
{
  __shared__ float zt[4][16][16];
  const int wave = threadIdx.x >> 5;
  const int lane = threadIdx.x & 31;
  const int half = lane >> 4, l16 = lane & 15;
  const int cg = blockIdx.x;          // H column group
  const int m0 = blockIdx.y * 16;     // batch tile
  const int n0 = wave * H_ + cg * 16; // column in 4H (gate = wave)

  v8f_t acc = {0.f, 0.f, 0.f, 0.f, 0.f, 0.f, 0.f, 0.f};
  acc = gemm_wave_full(xseq + (long)t * IN_, (long)T_ * IN_, Wx, 4 * H_, IN_, m0, n0, acc);
  acc = gemm_wave_full(reads, R_ * W_, Wx + (long)IN_ * 4 * H_, 4 * H_, R_ * W_, m0, n0, acc);
  acc = gemm_wave_full(h_prev, H_, Wh, 4 * H_, H_, m0, n0, acc);

  const float bias = bl[n0 + l16];
#pragma unroll
  for (int v = 0; v < 8; v++) zt[wave][v + 8 * half][l16] = acc[v] + bias;
  __syncthreads();

  for (int e = threadIdx.x; e < 256; e += 128) {
    int m = e >> 4, j = e & 15;
    float ig = zt[0][m][j], fg = zt[1][m][j], gg = zt[2][m][j], og = zt[3][m][j];
    int b = m0 + m, col = cg * 16 + j;
    float cold = c[b * H_ + col];
    float cn = sigmoidf_(fg) * cold + sigmoidf_(ig) * tanhf(gg);
    float hn = sigmoidf_(og) * tanhf(cn);
    c[b * H_ + col] = cn;
    h_next[b * H_ + col] = hn;
  }
}

// ---------------------------------------------------------------------------
// Interface: v = h @ W_iface + b_iface  ([32x512]@[512x471], padded to 480)
// grid 15, 128 threads = 4 waves; tile = blockIdx.x*4 + wave in [0,60)
// ---------------------------------------------------------------------------
__global__ void __launch_bounds__(128) k_iface(const float* __restrict__ h,
                                               const float* __restrict__ Wi,
                                               const float* __restrict__ bi,
                                               float* __restrict__ vout)
{
  const int wave = threadIdx.x >> 5;
  const int lane = threadIdx.x & 31;
  const int half = lane >> 4, l16 = lane & 15;
  const int tile = blockIdx.x * 4 + wave; // 0..59
  const int bm = tile / 30, ct = tile % 30;
  const int m0 = bm * 16, n0 = ct * 16;

  v8f_t acc = {0.f, 0.f, 0.f, 0.f, 0.f, 0.f, 0.f, 0.f};
  acc = gemm_wave_guard(h, H_, Wi, IFACE_, H_, m0, n0, IFACE_, acc);

  int col = n0 + l16;
  if (col < IFACE_) {
    float bias = bi[col];
#pragma unroll
    for (int v = 0; v < 8; v++) vout[(long)(m0 + v + 8 * half) * VSTRIDE_ + col] = acc[v] + bias;
  }
}

// ---------------------------------------------------------------------------
// Per-example memory op kernel: one 256-thread workgroup per batch example.
// M tile (128x64 f32, 32 KB) staged LDS<->global via the Tensor Data Mover,
// overlapped with interface-vector parsing by the other waves.
// ---------------------------------------------------------------------------
__global__ void __launch_bounds__(256) k_mem(const float* __restrict__ vbuf,
                                             float* __restrict__ Mg,
                                             float* __restrict__ usage_g,
                                             float* __restrict__ link_g,
                                             float* __restrict__ prec_g,
                                             float* __restrict__ ww_g,
                                             float* __restrict__ wr_g,
                                             float* __restrict__ reads_g)
{
  __shared__ float Msh[N_ * W_];       // 32 KB memory matrix
  __shared__ float wr_old[R_ * N_];    // old read weights (reused for new at end)
  __shared__ float ww_new[N_];
  __shared__ float prec_old[N_];
  __shared__ float usage_new[N_];
  __shared__ float alloc_a[N_];
  __shared__ float cwv[N_];
  __shared__ float sims[R_ * N_];
  __shared__ float fwv[R_ * N_];
  __shared__ float bwv[R_ * N_];
  __shared__ float rkeys[R_ * W_];
  __shared__ float wkey[W_];
  __shared__ float ers[W_];
  __shared__ float wvec[W_];
  __shared__ float rbeta[R_], freeg[R_], modes_s[R_ * 3], knr[R_];
  __shared__ float scal[8];
  __shared__ float srt_u[N_];
  __shared__ int srt_i[N_];
  __shared__ float Mnorm[N_];

  const int b = blockIdx.x;
  const int tid = threadIdx.x;
  const int wave = tid >> 5, lane = tid & 31;
  const float* vb = vbuf + (long)b * VSTRIDE_;

  // ---- async: TDM DMA of M[b] (128x64 f32) into LDS, issued by wave 0 ----
  if (wave == 0) {
    tdm_tile_2d_f32(false, (void*)Msh, (void*)(Mg + (long)b * N_ * W_),
                    W_, N_, W_, N_, W_);
  }

  // ---- parse interface vector (overlaps with the DMA) ----
  rkeys[tid] = vb[tid]; // [0,256): R*W read keys, blockDim==256
  for (int i = tid; i < W_; i += 256) {
    wkey[i] = vb[260 + i];              // write key
    ers[i]  = sigmoidf_(vb[325 + i]);   // erase
    wvec[i] = vb[389 + i];              // write vector
  }
  if (tid < R_) {
    rbeta[tid] = 1.0f + softplusf_(vb[256 + tid]);
    freeg[tid] = sigmoidf_(vb[453 + tid]);
    float a0 = vb[459 + tid * 3], a1 = vb[460 + tid * 3], a2 = vb[461 + tid * 3];
    float mx = fmaxf(a0, fmaxf(a1, a2));
    float e0 = expf(a0 - mx), e1 = expf(a1 - mx), e2 = expf(a2 - mx);
    float s = e0 + e1 + e2;
    modes_s[tid * 3 + 0] = e0 / s;
    modes_s[tid * 3 + 1] = e1 / s;
    modes_s[tid * 3 + 2] = e2 / s;
  }
  if (tid == 0) {
    scal[0] = 1.0f + softplusf_(vb[324]); // write_beta
    scal[1] = sigmoidf_(vb[457]);         // alloc_g
    scal[2] = sigmoidf_(vb[458]);         // write_g
  }
  // ---- load small state vectors ----
  for (int i = tid; i < R_ * N_; i += 256) wr_old[i] = wr_g[(long)b * R_ * N_ + i];
  if (tid < N_) prec_old[tid] = prec_g[b * N_ + tid];

  if (wave == 0) __builtin_amdgcn_s_wait_tensorcnt(0); // M DMA complete
  __syncthreads();

  // ---- usage update ----
  if (tid < N_) {
    float u = usage_g[b * N_ + tid];
    float w = ww_g[b * N_ + tid];
    float ret = 1.0f;
#pragma unroll
    for (int r = 0; r < R_; r++) ret *= (1.0f - freeg[r] * wr_old[r * N_ + tid]);
    float un = (u + w - u * w) * ret;
    usage_new[tid] = un;
    srt_u[tid] = un;
    srt_i[tid] = tid;
  }
  __syncthreads();

  // ---- bitonic sort ascending on (usage, index) — stable like jnp.argsort ----
  for (int k = 2; k <= N_; k <<= 1) {
    for (int j = k >> 1; j > 0; j >>= 1) {
      if (tid < N_) {
        int ixj = tid ^ j;
        if (ixj > tid) {
          float u1 = srt_u[tid], u2 = srt_u[ixj];
          int i1 = srt_i[tid], i2 = srt_i[ixj];
          bool gt = (u1 > u2) || (u1 == u2 && i1 > i2);
          bool up = ((tid & k) == 0);
          if (gt == up) {
            srt_u[tid] = u2; srt_u[ixj] = u1;
            srt_i[tid] = i2; srt_i[ixj] = i1;
          }
        }
      }
      __syncthreads();
    }
  }
  // ---- allocation weighting (serial cumprod over 128 by thread 0) ----
  if (tid == 0) {
    float cp = 1.0f;
    for (int k = 0; k < N_; k++) {
      float su = srt_u[k];
      alloc_a[srt_i[k]] = (1.0f - su) * cp;
      cp *= su;
    }
  }

  // ---- write content addressing on OLD M (wave-per-row) ----
  for (int n = wave; n < N_; n += 8) {
    float nrm = 0.f, dt = 0.f;
    for (int e = lane; e < W_; e += 32) {
      float m = Msh[n * W_ + e];
      nrm += m * m;
      dt += m * wkey[e];
    }
#pragma unroll
    for (int off = 16; off > 0; off >>= 1) {
      nrm += __shfl_xor(nrm, off, 32);
      dt += __shfl_xor(dt, off, 32);
    }
    if (lane == 0) { Mnorm[n] = sqrtf(nrm); cwv[n] = dt; }
  }
  if (wave == 0) {
    float kn = 0.f;
    for (int e = lane; e < W_; e += 32) { float x = wkey[e]; kn += x * x; }
#pragma unroll
    for (int off = 16; off > 0; off >>= 1) kn += __shfl_xor(kn, off, 32);
    if (lane == 0) scal[3] = sqrtf(kn) + EPS_;
  }
  __syncthreads();
  if (tid < N_) cwv[tid] = scal[0] * cwv[tid] / ((Mnorm[tid] + EPS_) * scal[3]);
  __syncthreads();
  if (tid == 0) { // softmax over N=128 (serial, tiny)
    float mx = -3.0e38f;
    for (int n = 0; n < N_; n++) mx = fmaxf(mx, cwv[n]);
    float s = 0.f;
    for (int n = 0; n < N_; n++) { float e = expf(cwv[n] - mx); cwv[n] = e; s += e; }
    float inv = 1.0f / s;
    for (int n = 0; n < N_; n++) cwv[n] *= inv;
  }
  __syncthreads();

  // ---- write weighting ----
  if (tid < N_) {
    float ag = scal[1], wg = scal[2];
    ww_new[tid] = wg * (ag * alloc_a[tid] + (1.0f - ag) * cwv[tid]);
  }
  __syncthreads();
  if (tid == 0) {
    float s = 0.f;
    for (int n = 0; n < N_; n++) s += ww_new[n];
    scal[6] = s; // sum(ww_n)
  }
  // ---- memory write: M_n = M*(1 - ww*erase) + ww*write_vec ----
  for (int i = tid; i < N_ * W_; i += 256) {
    int n = i >> 6, e = i & 63;
    float wwn = ww_new[n];
    Msh[i] = Msh[i] * (1.0f - wwn * ers[e]) + wwn * wvec[e];
  }
  for (int i = tid; i < R_ * N_; i += 256) { fwv[i] = 0.f; bwv[i] = 0.f; }
  __syncthreads();

  // ---- precedence ----
  if (tid < N_) prec_g[b * N_ + tid] = (1.0f - scal[6]) * prec_old[tid] + ww_new[tid];

  // ---- link update (in place, streamed) + fused fw/bw matvecs ----
  {
    float bwp[R_][4] = {};
    float* lb = link_g + (long)b * N_ * N_;
    for (int i = wave; i < N_; i += 8) {
      float wwi = ww_new[i];
      float fwp[R_] = {0.f, 0.f, 0.f, 0.f};
      float wri[R_];
#pragma unroll
      for (int r = 0; r < R_; r++) wri[r] = wr_old[r * N_ + i];
#pragma unroll
      for (int jj = 0; jj < 4; jj++) {
        int j = lane + jj * 32;
        float l = lb[i * N_ + j];
        float ln = (i == j) ? 0.f : (1.0f - wwi - ww_new[j]) * l + wwi * prec_old[j];
        lb[i * N_ + j] = ln;
#pragma unroll
        for (int r = 0; r < R_; r++) {
          fwp[r] += ln * wr_old[r * N_ + j]; // fw[r][i] += L[i][j]*wr[r][j]
          bwp[r][jj] += ln * wri[r];         // bw[r][j] += L[i][j]*wr[r][i]
        }
      }
#pragma unroll
      for (int r = 0; r < R_; r++) {
        float f = fwp[r];
#pragma unroll
        for (int off = 16; off > 0; off >>= 1) f += __shfl_xor(f, off, 32);
        if (lane == 0) fwv[r * N_ + i] = f;
      }
    }
#pragma unroll
    for (int r = 0; r < R_; r++)
#pragma unroll
      for (int jj = 0; jj < 4; jj++) atomicAdd(&bwv[r * N_ + lane + jj * 32], bwp[r][jj]);
  }
  __syncthreads();

  // ---- read content addressing on NEW M ----
  for (int n = wave; n < N_; n += 8) {
    float a0 = 0.f, a1 = 0.f, a2 = 0.f, a3 = 0.f, nr = 0.f;
    for (int e = lane; e < W_; e += 32) {
      float m = Msh[n * W_ + e];
      nr += m * m;
      a0 += m * rkeys[0 * W_ + e];
      a1 += m * rkeys[1 * W_ + e];
      a2 += m * rkeys[2 * W_ + e];
      a3 += m * rkeys[3 * W_ + e];
    }
#pragma unroll
    for (int off = 16; off > 0; off >>= 1) {
      nr += __shfl_xor(nr, off, 32);
      a0 += __shfl_xor(a0, off, 32);
      a1 += __shfl_xor(a1, off, 32);
      a2 += __shfl_xor(a2, off, 32);
      a3 += __shfl_xor(a3, off, 32);
    }
    if (lane == 0) {
      Mnorm[n] = sqrtf(nr);
      sims[0 * N_ + n] = a0; sims[1 * N_ + n] = a1;
      sims[2 * N_ + n] = a2; sims[3 * N_ + n] = a3;
    }
  }
  if (wave < R_) {
    float kn = 0.f;
    for (int e = lane; e < W_; e += 32) { float x = rkeys[wave * W_ + e]; kn += x * x; }
#pragma unroll
    for (int off = 16; off > 0; off >>= 1) kn += __shfl_xor(kn, off, 32);
    if (lane == 0) knr[wave] = sqrtf(kn) + EPS_;
  }
  __syncthreads();
  for (int i = tid; i < R_ * N_; i += 256) {
    int r = i >> 7, n = i & 127;
    sims[i] = rbeta[r] * sims[i] / ((Mnorm[n] + EPS_) * knr[r]);
  }
  __syncthreads();
  if (tid < R_) { // per-key softmax over N
    int r = tid;
    float mx = -3.0e38f;
    for (int n = 0; n < N_; n++) mx = fmaxf(mx, sims[r * N_ + n]);
    float s = 0.f;
    for (int n = 0; n < N_; n++) { float e = expf(sims[r * N_ + n] - mx); sims[r * N_ + n] = e; s += e; }
    float inv = 1.0f / s;
    for (int n = 0; n < N_; n++) sims[r * N_ + n] *= inv;
  }
  __syncthreads();

  // ---- new read weights: wr = m0*bw + m1*cr + m2*fw ----
  for (int i = tid; i < R_ * N_; i += 256) {
    int r = i >> 7;
    float w = modes_s[r * 3 + 0] * bwv[i] + modes_s[r * 3 + 1] * sims[i] + modes_s[r * 3 + 2] * fwv[i];
    wr_g[(long)b * R_ * N_ + i] = w;
    wr_old[i] = w; // reuse slot as new weights
  }
  __syncthreads();

  // ---- reads: reads[r][e] = sum_n wr_n[r][n] * M_n[n][e] ----
  {
    int r = tid >> 6, e = tid & 63;
    float acc = 0.f;
    for (int n = 0; n < N_; n++) acc += wr_old[r * N_ + n] * Msh[n * W_ + e];
    reads_g[(long)b * R_ * W_ + tid] = acc;
  }

  // ---- write back state; M via TDM store from LDS ----
  if (tid < N_) {
    usage_g[b * N_ + tid] = usage_new[tid];
    ww_g[b * N_ + tid] = ww_new[tid];
  }
  __syncthreads(); // all Msh updates done before DMA-out
  if (wave == 0) {
    tdm_tile_2d_f32(true, (void*)Msh, (void*)(Mg + (long)b * N_ * W_),
                    W_, N_, W_, N_, W_);
    __builtin_amdgcn_s_wait_tensorcnt(0);
  }
}

// ---------------------------------------------------------------------------
// Output: out[:, t, :] = [reads, h] @ W_out + b_out  ([32x768]@[768x512])
// grid 16, 128 threads = 4 waves; 64 tiles.
// ---------------------------------------------------------------------------
__global__ void __launch_bounds__(128) k_out(const float* __restrict__ reads,
                                             const float* __restrict__ h,
                                             const float* __restrict__ Wo,
                                             const float* __restrict__ bo,
                                             float* __restrict__ out, int t)
{
  const int wave = threadIdx.x >> 5;
  const int lane = threadIdx.x & 31;
  const int half = lane >> 4, l16 = lane & 15;
  const int tile = blockIdx.x * 4 + wave; // 0..63
  const int bm = tile / 32, ct = tile % 32;
  const int m0 = bm * 16, n0 = ct * 16;

  v8f_t acc = {0.f, 0.f, 0.f, 0.f, 0.f, 0.f, 0.f, 0.f};
  acc = gemm_wave_full(reads, R_ * W_, Wo, OUT_, R_ * W_, m0, n0, acc);
  acc = gemm_wave_full(h, H_, Wo + (long)(R_ * W_) * OUT_, OUT_, H_, m0, n0, acc);

  const int col = n0 + l16;
  const float bias = bo[col];
  float* D = out + (long)t * OUT_; // out[b][t][o] = out[(b*T + t)*OUT + o]
#pragma unroll
  for (int v = 0; v < 8; v++)
    D[(long)(m0 + v + 8 * half) * ((long)T_ * OUT_) + col] = acc[v] + bias;
}

// ---------------------------------------------------------------------------
// Host launcher: 128 sequential steps, 4 kernels each, all on `stream`.
// ---------------------------------------------------------------------------
extern "C" void kernel_launch(void* const* d_in, const int* in_sizes, int n_in,
                              void* d_out, int out_size, void* d_ws, size_t ws_size,
                              hipStream_t stream)
{
  (void)in_sizes; (void)n_in; (void)out_size; (void)ws_size;
  const float* xseq   = (const float*)d_in[0]; // [B,T,IN]
  const float* Wx     = (const float*)d_in[1]; // [768, 2048]
  const float* Wh     = (const float*)d_in[2]; // [512, 2048]
  const float* b_lstm = (const float*)d_in[3]; // [2048]
  const float* Wi     = (const float*)d_in[4]; // [512, 471]
  const float* bi     = (const float*)d_in[5]; // [471]
  const float* Wo     = (const float*)d_in[6]; // [768, 512]
  const float* bo     = (const float*)d_in[7]; // [512]
  float* out = (float*)d_out;                  // [B,T,OUT]

  // Workspace layout (floats)
  float* ws = (float*)d_ws;
  float* h0    = ws;                   // 16384
  float* h1    = h0 + B_ * H_;         // 16384
  float* c     = h1 + B_ * H_;         // 16384
  float* M     = c + B_ * H_;          // 262144
  float* usage = M + B_ * N_ * W_;     // 4096
  float* link  = usage + B_ * N_;      // 524288
  float* prec  = link + B_ * N_ * N_;  // 4096
  float* ww    = prec + B_ * N_;       // 4096
  float* wr    = ww + B_ * N_;         // 16384
  float* reads = wr + B_ * R_ * N_;    // 8192
  float* vbuf  = reads + B_ * R_ * W_; // 32*480

  const int state_n = (int)(vbuf - ws); // everything before vbuf must start at 0
  k_zero<<<(state_n + 255) / 256, 256, 0, stream>>>(ws, state_n);

  for (int t = 0; t < T_; t++) {
    const float* hp = (t & 1) ? h1 : h0;
    float* hn       = (t & 1) ? h0 : h1;
    k_lstm<<<dim3(H_ / 16, B_ / 16), 128, 0, stream>>>(xseq, t, Wx, Wh, b_lstm, reads, hp, hn, c);
    k_iface<<<15, 128, 0, stream>>>(hn, Wi, bi, vbuf);
    k_mem<<<B_, 256, 0, stream>>>(vbuf, M, usage, link, prec, ww, wr, reads);
    k_out<<<16, 128, 0, stream>>>(reads, hn, Wo, bo, out, t);
  }
}